// GCN_26723286515820
// MI455X (gfx1250) — compile-verified
//
#include <hip/hip_runtime.h>

typedef float v2f __attribute__((ext_vector_type(2)));
typedef float v8f __attribute__((ext_vector_type(8)));

#define DIN  128
#define DH   128
#define DOUT 40
#define DOUTP 48
#define BN_EPS 1e-5f

__device__ __forceinline__ void atomAddF(float* p, float v) {
    // hardware global_atomic_add_f32 (no CAS loop)
    unsafeAtomicAdd(p, v);
}

// ---------------------------------------------------------------- init
__global__ __launch_bounds__(256) void k_init(float* __restrict__ deg,
                                              float* __restrict__ colsum,
                                              float* __restrict__ colsumsq,
                                              int n) {
    int i = blockIdx.x * 256 + threadIdx.x;
    if (i < n) deg[i] = 1.0f;            // self loop
    if (i < DH) { colsum[i] = 0.f; colsumsq[i] = 0.f; }
}

// ---------------------------------------------------------------- degree
__global__ __launch_bounds__(256) void k_degree(const int* __restrict__ dst,
                                                float* __restrict__ deg, int E) {
    size_t stride = (size_t)gridDim.x * 256;
    for (size_t e = (size_t)blockIdx.x * 256 + threadIdx.x; e < (size_t)E; e += stride)
        atomAddF(&deg[dst[e]], 1.0f);
}

__global__ __launch_bounds__(256) void k_dinv(const float* __restrict__ deg,
                                              float* __restrict__ dinv, int n) {
    int i = blockIdx.x * 256 + threadIdx.x;
    if (i < n) { float d = deg[i]; dinv[i] = (d > 0.f) ? rsqrtf(d) : 0.f; }
}

// ---------------------------------------------------------------- GEMM1: X1 = A(n x128) @ W1(128x128)
// one block = one 16-row M tile; 8 waves = 8 column tiles of 16
__global__ __launch_bounds__(256) void k_gemm1(const float* __restrict__ A,
                                               const float* __restrict__ W,
                                               float* __restrict__ X1, int n) {
    __shared__ float As[16 * 132];          // padded vs bank conflicts
    __shared__ float Ws[32 * 128];
    const int t    = threadIdx.x;
    const int m0   = blockIdx.x * 16;
    const int lane = t & 31, warp = t >> 5;
    const int l15  = lane & 15, hi = lane >> 4;

    for (int i = 0; i < 8; ++i) {           // 16x128 A tile, coalesced
        int idx = i * 256 + t;
        int r = idx >> 7, c = idx & 127;
        int gr = m0 + r;
        As[r * 132 + c] = (gr < n) ? A[(size_t)gr * DIN + c] : 0.f;
    }

    v8f acc = {};
    for (int kc = 0; kc < 128; kc += 32) {
        __syncthreads();
        for (int i = 0; i < 16; ++i) {      // 32x128 W chunk
            int idx = i * 256 + t;
            int kr = idx >> 7, c = idx & 127;
            Ws[kr * 128 + c] = W[(size_t)(kc + kr) * DH + c];
        }
        __syncthreads();
        for (int kk = 0; kk < 32; kk += 4) {
            v2f a, b;
            int ak = kc + kk + 2 * hi;      // ISA 32-bit A 16x4 layout
            a.x = As[l15 * 132 + ak];
            a.y = As[l15 * 132 + ak + 1];
            int bk = kk + 2 * hi;
            b.x = Ws[bk * 128 + warp * 16 + l15];
            b.y = Ws[(bk + 1) * 128 + warp * 16 + l15];
            acc = __builtin_amdgcn_wmma_f32_16x16x4_f32(false, a, false, b,
                                                        (short)0, acc, false, false);
        }
    }
    const int col = warp * 16 + l15;
    for (int v = 0; v < 8; ++v) {
        int r = m0 + v + 8 * hi;
        if (r < n) X1[(size_t)r * DH + col] = acc[v];
    }
}

// ------------------------------------------------- self-loop init of AG = X1 * dinv^2
__global__ __launch_bounds__(256) void k_self1(const float* __restrict__ X1,
                                               const float* __restrict__ dinv,
                                               float* __restrict__ AG, int n) {
    size_t tid = (size_t)blockIdx.x * 256 + threadIdx.x;
    if (tid >= (size_t)n * 32) return;
    size_t node = tid >> 5; int q = (int)(tid & 31);
    float s = dinv[node]; s *= s;
    float4 v = *((const float4*)(X1 + node * DH) + q);
    float4 o = make_float4(v.x * s, v.y * s, v.z * s, v.w * s);
    *((float4*)(AG + node * DH) + q) = o;
}

// ------------------------------------------------- edge scatter, 128 feats, wave/edge
__global__ __launch_bounds__(256) void k_agg1(const int* __restrict__ src,
                                              const int* __restrict__ dst,
                                              const float* __restrict__ dinv,
                                              const float* __restrict__ X1,
                                              float* __restrict__ AG, int E) {
    const int lane = threadIdx.x & 31;
    size_t gw = (size_t)blockIdx.x * 8 + (threadIdx.x >> 5);
    size_t nw = (size_t)gridDim.x * 8;
    for (size_t e = gw; e < (size_t)E; e += nw) {
        int s = src[e], d = dst[e];                 // wave-uniform -> scalar loads
        float w = dinv[s] * dinv[d];
        float4 xv = *((const float4*)(X1 + (size_t)s * DH) + lane);
        float* o = AG + (size_t)d * DH + lane * 4;
        atomAddF(o + 0, xv.x * w);
        atomAddF(o + 1, xv.y * w);
        atomAddF(o + 2, xv.z * w);
        atomAddF(o + 3, xv.w * w);
    }
}

// ------------------------------------------------- BN column statistics
__global__ __launch_bounds__(256) void k_stats(const float* __restrict__ AG,
                                               float* __restrict__ colsum,
                                               float* __restrict__ colsumsq, int n) {
    int col = threadIdx.x & 127;
    int rs  = blockIdx.x * 2 + (threadIdx.x >> 7);
    int stride = gridDim.x * 2;
    float s = 0.f, q = 0.f;
    for (int r = rs; r < n; r += stride) {
        float v = AG[(size_t)r * DH + col];
        s += v; q += v * v;
    }
    atomAddF(&colsum[col], s);
    atomAddF(&colsumsq[col], q);
}

__global__ __launch_bounds__(128) void k_bnparams(const float* __restrict__ colsum,
                                                  const float* __restrict__ colsumsq,
                                                  const float* __restrict__ gamma,
                                                  float* __restrict__ meanA,
                                                  float* __restrict__ scaleA, int n) {
    int c = threadIdx.x;
    float inv_n = 1.0f / (float)n;
    float m = colsum[c] * inv_n;
    float var = colsumsq[c] * inv_n - m * m;
    var = fmaxf(var, 0.f);
    meanA[c]  = m;
    scaleA[c] = gamma[c] * rsqrtf(var + BN_EPS);
}

// ------------------------------------------------- GEMM2 with fused BN+ReLU on A-load
// one block = one 16-row tile; 4 waves, waves 0..2 cover 48 padded cols
__global__ __launch_bounds__(128) void k_gemm2(const float* __restrict__ AG,
                                               const float* __restrict__ W2,
                                               const float* __restrict__ meanA,
                                               const float* __restrict__ scaleA,
                                               const float* __restrict__ beta,
                                               float* __restrict__ X2, int n) {
    __shared__ float As[16 * 132];
    __shared__ float Ws[128 * DOUTP];
    const int t    = threadIdx.x;
    const int m0   = blockIdx.x * 16;
    const int lane = t & 31, warp = t >> 5;
    const int l15  = lane & 15, hi = lane >> 4;

    for (int i = 0; i < 16; ++i) {
        int idx = i * 128 + t;
        int r = idx >> 7, c = idx & 127;
        int gr = m0 + r;
        float v = 0.f;
        if (gr < n) {
            v = AG[(size_t)gr * DH + c];
            v = (v - meanA[c]) * scaleA[c] + beta[c];   // BN (b1 cancels)
            v = fmaxf(v, 0.f);                          // ReLU
        }
        As[r * 132 + c] = v;
    }
    for (int i = 0; i < 48; ++i) {                      // 128x48 zero-padded W2
        int idx = i * 128 + t;
        int k = idx / DOUTP, c = idx - k * DOUTP;
        Ws[idx] = (c < DOUT) ? W2[(size_t)k * DOUT + c] : 0.f;
    }
    __syncthreads();
    if (warp >= 3) return;

    v8f acc = {};
    for (int k0 = 0; k0 < 128; k0 += 4) {
        v2f a, b;
        int ak = k0 + 2 * hi;
        a.x = As[l15 * 132 + ak];
        a.y = As[l15 * 132 + ak + 1];
        b.x = Ws[ak * DOUTP + warp * 16 + l15];
        b.y = Ws[(ak + 1) * DOUTP + warp * 16 + l15];
        acc = __builtin_amdgcn_wmma_f32_16x16x4_f32(false, a, false, b,
                                                    (short)0, acc, false, false);
    }
    const int col = warp * 16 + l15;
    if (col < DOUT) {
        for (int v = 0; v < 8; ++v) {
            int r = m0 + v + 8 * hi;
            if (r < n) X2[(size_t)r * DOUT + col] = acc[v];
        }
    }
}

// ------------------------------------------------- out = X2*dinv^2 + b2 (self loop + bias)
__global__ __launch_bounds__(256) void k_self2(const float* __restrict__ X2,
                                               const float* __restrict__ dinv,
                                               const float* __restrict__ b2,
                                               float* __restrict__ out, int n) {
    size_t idx = (size_t)blockIdx.x * 256 + threadIdx.x;
    if (idx >= (size_t)n * DOUT) return;
    size_t node = idx / DOUT; int c = (int)(idx - node * DOUT);
    float s = dinv[node]; s *= s;
    out[idx] = X2[idx] * s + b2[c];
}

// ------------------------------------------------- edge scatter, 40 feats, wave/edge
__global__ __launch_bounds__(256) void k_agg2(const int* __restrict__ src,
                                              const int* __restrict__ dst,
                                              const float* __restrict__ dinv,
                                              const float* __restrict__ X2,
                                              float* __restrict__ out, int E) {
    const int lane = threadIdx.x & 31;
    size_t gw = (size_t)blockIdx.x * 8 + (threadIdx.x >> 5);
    size_t nw = (size_t)gridDim.x * 8;
    for (size_t e = gw; e < (size_t)E; e += nw) {
        int s = src[e], d = dst[e];
        float w = dinv[s] * dinv[d];
        const float* xs = X2 + (size_t)s * DOUT;
        float* od = out + (size_t)d * DOUT;
        atomAddF(od + lane, xs[lane] * w);
        if (lane < DOUT - 32)
            atomAddF(od + 32 + lane, xs[32 + lane] * w);
    }
}

// =================================================================
extern "C" void kernel_launch(void* const* d_in, const int* in_sizes, int n_in,
                              void* d_out, int out_size, void* d_ws, size_t ws_size,
                              hipStream_t stream) {
    const float* node_feat = (const float*)d_in[0];
    const float* W1        = (const float*)d_in[1];
    // b1 = d_in[2] cancels exactly inside BatchNorm (shift removed by mean)
    const float* gamma1    = (const float*)d_in[3];
    const float* beta1     = (const float*)d_in[4];
    const float* W2        = (const float*)d_in[5];
    const float* b2        = (const float*)d_in[6];
    const int*   edge      = (const int*)d_in[7];

    const int n = in_sizes[0] / DIN;       // 50000
    const int E = in_sizes[7] / 2;         // 800000
    const int* src = edge;
    const int* dst = edge + E;

    float* wsf      = (float*)d_ws;
    float* deg      = wsf;                              // n
    float* dinv     = deg + n;                          // n
    float* x1       = dinv + n;                         // n*128
    float* ag       = x1 + (size_t)n * DH;              // n*128
    float* x2       = ag + (size_t)n * DH;              // n*40
    float* colsum   = x2 + (size_t)n * DOUT;            // 128
    float* colsumsq = colsum + DH;                      // 128
    float* meanA    = colsumsq + DH;                    // 128
    float* scaleA   = meanA + DH;                       // 128
    float* out      = (float*)d_out;

    const int mtiles = (n + 15) / 16;

    k_init  <<<(n + 255) / 256, 256, 0, stream>>>(deg, colsum, colsumsq, n);
    k_degree<<<2048, 256, 0, stream>>>(dst, deg, E);
    k_dinv  <<<(n + 255) / 256, 256, 0, stream>>>(deg, dinv, n);

    k_gemm1 <<<mtiles, 256, 0, stream>>>(node_feat, W1, x1, n);
    k_self1 <<<((size_t)n * 32 + 255) / 256, 256, 0, stream>>>(x1, dinv, ag, n);
    k_agg1  <<<4096, 256, 0, stream>>>(src, dst, dinv, x1, ag, E);

    k_stats <<<512, 256, 0, stream>>>(ag, colsum, colsumsq, n);
    k_bnparams<<<1, 128, 0, stream>>>(colsum, colsumsq, gamma1, meanA, scaleA, n);

    k_gemm2 <<<mtiles, 128, 0, stream>>>(ag, W2, meanA, scaleA, beta1, x2, n);
    k_self2 <<<((size_t)n * DOUT + 255) / 256, 256, 0, stream>>>(x2, dinv, b2, out, n);
    k_agg2  <<<4096, 256, 0, stream>>>(src, dst, dinv, x2, out, E);
}